// ResourceArea_60043642798477
// MI455X (gfx1250) — compile-verified
//
#include <hip/hip_runtime.h>

// ---------------- problem constants (match reference) ----------------
constexpr int   NBX      = 512;
constexpr int   NBY      = 512;
constexpr int   NB       = NBX * NBY;        // 262144 bins
constexpr int   NCKSR    = 48;
constexpr int   NCE      = 96;
constexpr int   KW       = 5;
constexpr float HALF     = 1.5f;             // STDDEV_TRUNC * STDDEV = 2.5*0.6
constexpr float INV_SDS2 = 1.17851130198f;   // 1/(0.6*sqrt(2))

// workspace layout (float units), SoA [slot][bin] for ck/ce
//   [0, NB)                      lut_dem
//   [NB, NB + 48*NB)             ck_dem   (slot-major)
//   [49*NB, 49*NB + 96*NB)       ce_dem   (slot-major)
//   [145*NB, 146*NB)             lut_scale
//   [146*NB, 147*NB)             ff_scale
constexpr size_t OFF_CK  = (size_t)NB;
constexpr size_t OFF_CE  = (size_t)NB * 49;
constexpr size_t OFF_LSC = (size_t)NB * 145;
constexpr size_t OFF_FSC = (size_t)NB * 146;
constexpr size_t ZERO_F  = (size_t)NB * 145; // floats that must start at zero

typedef float v2f __attribute__((ext_vector_type(2)));
typedef float v8f __attribute__((ext_vector_type(8)));

// Truncated-Gaussian bin-overlap weights along one axis (mirrors _axis_weights).
__device__ __forceinline__ void axis_weights(float pos, int nbins,
                                             int bins[KW], float w[KW]) {
    int   lo  = (int)floorf(pos - HALF);
    float wlo = pos - HALF;
    float whi = pos + HALF;
    float s   = 0.f;
#pragma unroll
    for (int k = 0; k < KW; ++k) {
        int   b   = lo + k;
        float elo = (float)b;
        float ehi = elo + 1.0f;
        float lc  = fminf(fmaxf(elo, wlo), whi);
        float hc  = fminf(fmaxf(ehi, wlo), whi);
        float ph  = 0.5f * (1.0f + erff((hc - pos) * INV_SDS2));
        float pl  = 0.5f * (1.0f + erff((lc - pos) * INV_SDS2));
        float ww  = ph - pl;
        bool valid = (b >= 0) && (b < nbins);
        ww = valid ? ww : 0.0f;
        s += ww;
        bins[k] = min(max(b, 0), nbins - 1);
        w[k]    = ww;
    }
    float r = 1.0f / (s + 1e-12f);
#pragma unroll
    for (int k = 0; k < KW; ++k) w[k] *= r;
}

// ---------------- phase 0: zero demand arrays ----------------
__global__ void zero_kernel(float4* __restrict__ p, size_t n4) {
    size_t stride = (size_t)gridDim.x * blockDim.x;
    for (size_t i = (size_t)blockIdx.x * blockDim.x + threadIdx.x; i < n4; i += stride)
        p[i] = make_float4(0.f, 0.f, 0.f, 0.f);
}

// ---------------- phase 1: scatter demand (f32 atomics, land in 192MB L2) ----
__global__ void scatter_kernel(const float* __restrict__ pos,
                               const int*   __restrict__ luts,
                               const int*   __restrict__ ffs,
                               const int*   __restrict__ ctrl,
                               float*       __restrict__ ws,
                               int n) {
    float* lut_dem = ws;
    float* ck_dem  = ws + OFF_CK;
    float* ce_dem  = ws + OFF_CE;
    int stride = gridDim.x * blockDim.x;
    for (int i = blockIdx.x * blockDim.x + threadIdx.x; i < n; i += stride) {
        __builtin_prefetch(pos + 2 * (i + stride), 0, 1);   // global_prefetch_b8
        __builtin_prefetch(luts + (i + stride), 0, 1);
        bool lut = luts[i] > 0;
        bool ff  = (!lut) && (ffs[i] > 0);
        if (!lut && !ff) continue;
        float px = pos[2 * i + 0];
        float py = pos[2 * i + 1];
        int bx[KW], by[KW];
        float wx[KW], wy[KW];
        axis_weights(px, NBX, bx, wx);
        axis_weights(py, NBY, by, wy);
        if (lut) {
#pragma unroll
            for (int kx = 0; kx < KW; ++kx) {
#pragma unroll
                for (int ky = 0; ky < KW; ++ky) {
                    float w = wx[kx] * wy[ky];
                    if (w > 0.f)
                        atomicAdd(&lut_dem[bx[kx] * NBY + by[ky]], w);
                }
            }
        } else {
            int cks = ctrl[2 * i + 0];
            int cev = ctrl[2 * i + 1];
            float* ckp = ck_dem + (size_t)cks * NB;
            float* cep = ce_dem + (size_t)cev * NB;
#pragma unroll
            for (int kx = 0; kx < KW; ++kx) {
#pragma unroll
                for (int ky = 0; ky < KW; ++ky) {
                    float w = wx[kx] * wy[ky];
                    if (w > 0.f) {
                        int idx = bx[kx] * NBY + by[ky];
                        atomicAdd(&ckp[idx], w);
                        atomicAdd(&cep[idx], w);
                    }
                }
            }
        }
    }
}

// ---------------- phase 2: per-bin reduction via V_WMMA_F32_16X16X4_F32 -----
// One wave32 handles 16 bins. A = 16(bins) x 4(slots) f32 per documented layout
// (lane m = lane&15 holds K-pair selected by lane>>4). B = all-ones, which is
// layout-independent and turns WMMA into a row-sum: D[m][n] = sum_k A[m][k]+C.
// Three accumulators: ceil(ck/8) sums, raw ck sums, ceil(ce/4) sums.
__global__ void reduce_kernel(const float* __restrict__ ws,
                              float* __restrict__ lut_scale,
                              float* __restrict__ ff_scale) {
    const float* lut_dem = ws;
    const float* ck_dem  = ws + OFF_CK;
    const float* ce_dem  = ws + OFF_CE;

    int wave  = (blockIdx.x * blockDim.x + threadIdx.x) >> 5;
    int lane  = threadIdx.x & 31;
    int base  = wave * 16;          // 16 bins per wave; grid sized exactly
    int m     = lane & 15;
    int khalf = lane >> 4;          // 0: K=0,1   1: K=2,3

    v2f ones = {1.0f, 1.0f};
    v8f acc_ck  = {};
    v8f acc_raw = {};
    v8f acc_ce  = {};

#pragma unroll
    for (int j = 0; j < NCKSR; j += 4) {
        int k0 = j + khalf * 2;
        float r0 = ck_dem[(size_t)(k0 + 0) * NB + base + m];
        float r1 = ck_dem[(size_t)(k0 + 1) * NB + base + m];
        v2f a_raw  = {r0, r1};
        v2f a_ceil = {ceilf(r0 * 0.125f), ceilf(r1 * 0.125f)};
        acc_ck  = __builtin_amdgcn_wmma_f32_16x16x4_f32(
                      false, a_ceil, false, ones, (short)0, acc_ck, false, false);
        acc_raw = __builtin_amdgcn_wmma_f32_16x16x4_f32(
                      false, a_raw, false, ones, (short)0, acc_raw, false, false);
    }
#pragma unroll
    for (int j = 0; j < NCE; j += 4) {
        int k0 = j + khalf * 2;
        float r0 = ce_dem[(size_t)(k0 + 0) * NB + base + m];
        float r1 = ce_dem[(size_t)(k0 + 1) * NB + base + m];
        v2f a_ceil = {ceilf(r0 * 0.25f), ceilf(r1 * 0.25f)};
        acc_ce = __builtin_amdgcn_wmma_f32_16x16x4_f32(
                     false, a_ceil, false, ones, (short)0, acc_ce, false, false);
    }

    // lut_scale: lanes 0..15 each own one bin
    if (lane < 16)
        lut_scale[base + m] = fmaxf(lut_dem[base + m] * (1.0f / 16.0f), 1.0f);

    // C/D layout: VGPR r -> M=r (lanes 0-15) / M=8+r (lanes 16-31); all N equal.
    if ((lane & 15) == 0) {
        int mb = base + khalf * 8;
#pragma unroll
        for (int r = 0; r < 8; ++r) {
            float eff_ck = 8.0f * acc_ck[r];
            float eff_ce = 4.0f * acc_ce[r];
            float eff    = fmaxf(eff_ck, eff_ce);
            float raw    = acc_raw[r];
            ff_scale[mb + r] = fmaxf(eff / fmaxf(raw, 1e-6f), 1.0f);
        }
    }
}

// ---------------- phase 3: gather areas ----------------
__global__ void gather_kernel(const float* __restrict__ pos,
                              const int*   __restrict__ luts,
                              const int*   __restrict__ ffs,
                              const float* __restrict__ lut_scale,
                              const float* __restrict__ ff_scale,
                              float*       __restrict__ out,
                              int n) {
    int stride = gridDim.x * blockDim.x;
    for (int i = blockIdx.x * blockDim.x + threadIdx.x; i < n; i += stride) {
        __builtin_prefetch(pos + 2 * (i + stride), 0, 1);
        bool lut = luts[i] > 0;
        bool ff  = (!lut) && (ffs[i] > 0);
        if (!lut && !ff) { out[i] = 0.0f; continue; }
        float px = pos[2 * i + 0];
        float py = pos[2 * i + 1];
        int bx[KW], by[KW];
        float wx[KW], wy[KW];
        axis_weights(px, NBX, bx, wx);
        axis_weights(py, NBY, by, wy);
        const float* sc = lut ? lut_scale : ff_scale;
        float area = 0.0f;
#pragma unroll
        for (int kx = 0; kx < KW; ++kx) {
#pragma unroll
            for (int ky = 0; ky < KW; ++ky) {
                float w = wx[kx] * wy[ky];
                area += w * sc[bx[kx] * NBY + by[ky]];
            }
        }
        out[i] = area;
    }
}

extern "C" void kernel_launch(void* const* d_in, const int* in_sizes, int n_in,
                              void* d_out, int out_size, void* d_ws, size_t ws_size,
                              hipStream_t stream) {
    const float* inst_pos = (const float*)d_in[0];
    const int*   luts     = (const int*)d_in[1];
    const int*   ffs      = (const int*)d_in[2];
    const int*   ctrl     = (const int*)d_in[3];
    float*       out      = (float*)d_out;
    float*       ws       = (float*)d_ws;
    const int n = in_sizes[1];           // N_INST

    // phase 0: zero the 145*NB demand floats (154 MB state -> lives in L2)
    zero_kernel<<<2048, 256, 0, stream>>>((float4*)ws, ZERO_F / 4);

    // phase 1: scatter demand
    scatter_kernel<<<2048, 256, 0, stream>>>(inst_pos, luts, ffs, ctrl, ws, n);

    // phase 2: per-bin reduction; exactly NB/16 = 16384 wave32 -> 2048x256
    reduce_kernel<<<NB / 16 / 8, 256, 0, stream>>>(ws, ws + OFF_LSC, ws + OFF_FSC);

    // phase 3: gather per-instance areas
    gather_kernel<<<2048, 256, 0, stream>>>(inst_pos, luts, ffs,
                                            ws + OFF_LSC, ws + OFF_FSC, out, n);
}